// EmbeddingLoss_32899449488181
// MI455X (gfx1250) — compile-verified
//
#include <hip/hip_runtime.h>

// Problem constants (fixed by the reference setup)
#define BB   16
#define DD   512
#define TT   2048
#define CC   20
#define KNUM 7
#define FF   140      // C*K
#define FP   144      // padded to 9 tiles of 16
#define NT   128      // T/16 column tiles
#define NCG  32       // NT/4 column groups of 4 tiles (k_bb)
#define NFT  9        // FP/16 fg tiles
#define EPSV 1e-8f
#define BIGV 1e9f

typedef float v2f __attribute__((ext_vector_type(2)));
typedef float v8f __attribute__((ext_vector_type(8)));

// Native f32 WMMA: D(16x16,f32) = A(16x4,f32) x B(4x16,f32) + C
__device__ __forceinline__ v8f wmma4(v2f a, v2f b, v8f c) {
  return __builtin_amdgcn_wmma_f32_16x16x4_f32(false, a, false, b, (short)0, c,
                                               false, false);
}

// ---- wave32 reductions (fixed order => deterministic) -----------------------
__device__ __forceinline__ float wred_min16(float v) {
  v = fminf(v, __shfl_xor(v, 1, 32));
  v = fminf(v, __shfl_xor(v, 2, 32));
  v = fminf(v, __shfl_xor(v, 4, 32));
  v = fminf(v, __shfl_xor(v, 8, 32));
  return v;
}
__device__ __forceinline__ float wred_max16(float v) {
  v = fmaxf(v, __shfl_xor(v, 1, 32));
  v = fmaxf(v, __shfl_xor(v, 2, 32));
  v = fmaxf(v, __shfl_xor(v, 4, 32));
  v = fmaxf(v, __shfl_xor(v, 8, 32));
  return v;
}
__device__ __forceinline__ float wred_sum32(float v) {
  v += __shfl_xor(v, 1, 32);
  v += __shfl_xor(v, 2, 32);
  v += __shfl_xor(v, 4, 32);
  v += __shfl_xor(v, 8, 32);
  v += __shfl_xor(v, 16, 32);
  return v;
}

// Order-preserving monotonic float<->uint encoding: atomicMin/Max on the
// encoding == exact float min/max (order-invariant => deterministic).
__device__ __forceinline__ unsigned fenc(float f) {
  unsigned u = __float_as_uint(f);
  return (u & 0x80000000u) ? ~u : (u | 0x80000000u);
}
__device__ __forceinline__ float fdec(unsigned u) {
  return (u & 0x80000000u) ? __uint_as_float(u & 0x7fffffffu)
                           : __uint_as_float(~u);
}

// ---- 0: init sentinels + fg padding ----------------------------------------
__global__ void k_init(unsigned* rowmax, unsigned* bgrmin, float* fgnrm,
                       float* fgm, int* fgidx) {
  int tid = blockIdx.x * 256 + threadIdx.x;
  if (tid < BB * TT) bgrmin[tid] = fenc(BIGV);
  if (tid < BB * FP) {
    rowmax[tid] = fenc(-BIGV);
    int f = tid % FP;
    if (f >= FF) { fgnrm[tid] = 1.f; fgm[tid] = 0.f; fgidx[tid] = 0; }
  }
}

// ---- 1: per-column L2 norms over D -----------------------------------------
__global__ void k_norm(const float* __restrict__ emb, float* __restrict__ nrm) {
  int t = blockIdx.x * 256 + threadIdx.x;
  int b = blockIdx.y;
  const float* p = emb + (size_t)b * DD * TT + t;
  float s = 0.f;
#pragma unroll 8
  for (int d = 0; d < DD; ++d) { float v = p[(size_t)d * TT]; s += v * v; }
  nrm[b * TT + t] = sqrtf(s);
}

// ---- 2: deterministic order-preserving bg compaction (block scan) ----------
__global__ void k_compact(const int* __restrict__ click, const int* clsnum,
                          int* __restrict__ bgidx, int* __restrict__ nbg) {
  int b = blockIdx.x, tid = threadIdx.x;
  __shared__ int s[256];
  __shared__ int base;
  if (tid == 0) base = 0;
  int cls = clsnum[0];
  __syncthreads();
  for (int c0 = 0; c0 < TT; c0 += 256) {
    int t = c0 + tid;
    int f = (click[b * TT + t] == cls) ? 1 : 0;
    s[tid] = f;
    __syncthreads();
    for (int off = 1; off < 256; off <<= 1) {       // inclusive scan
      int v = (tid >= off) ? s[tid - off] : 0;
      __syncthreads();
      s[tid] += v;
      __syncthreads();
    }
    if (f) bgidx[b * TT + base + s[tid] - 1] = t;
    __syncthreads();
    if (tid == 0) base += s[255];
    __syncthreads();
  }
  if (tid == 0) nbg[b] = base;
}

// ---- 3: top-K per (b,c), stable tie-break (lowest index) -------------------
__global__ void k_topk(const float* __restrict__ cas, const int* __restrict__ label,
                       const float* __restrict__ nrm, int* fgidx, float* fgnrm,
                       float* fgm) {
  int c = blockIdx.x, b = blockIdx.y, l = threadIdx.x;
  const float* row = cas + ((size_t)b * CC + c) * TT;
  float lm = (label[b * CC + c] == 1) ? 1.f : 0.f;
  int picked[KNUM];
  for (int k = 0; k < KNUM; ++k) {
    float bv = -3.0e38f; int bi = 0x7fffffff;
    for (int t = l; t < TT; t += 32) {
      bool skip = false;
      for (int j = 0; j < k; ++j) if (picked[j] == t) skip = true;
      if (skip) continue;
      float v = row[t];
      if (v > bv || (v == bv && t < bi)) { bv = v; bi = t; }
    }
    for (int off = 16; off >= 1; off >>= 1) {
      float ov = __shfl_xor(bv, off, 32);
      int   oi = __shfl_xor(bi, off, 32);
      if (ov > bv || (ov == bv && oi < bi)) { bv = ov; bi = oi; }
    }
    picked[k] = bi;
    if (l == 0) {
      int f = b * FP + c * KNUM + k;
      fgidx[f] = bi;
      fgnrm[f] = nrm[b * TT + bi];
      fgm[f]   = lm;
    }
  }
}

// ---- 4: gather fg embeddings to contiguous [FP x D] per batch --------------
__global__ void k_gather(const float* __restrict__ emb, const int* __restrict__ fgidx,
                         float* __restrict__ fgemb) {
  int f = blockIdx.x, b = blockIdx.y, tid = threadIdx.x;
  float* dst = fgemb + ((size_t)b * FP + f) * DD;
  if (f >= FF) { for (int d = tid; d < DD; d += 256) dst[d] = 0.f; return; }
  int idx = fgidx[b * FP + f];
  const float* src = emb + (size_t)b * DD * TT + idx;
  for (int d = tid; d < DD; d += 256) dst[d] = src[(size_t)d * TT];
}

// ---- 5: fg-fg gram: k-outer / 9 resident accumulators ----------------------
__global__ void k_ff(const float* __restrict__ fgemb, const float* __restrict__ fgnrm,
                     const float* __restrict__ fgm, float* __restrict__ part_ff) {
  int ti = blockIdx.x, b = blockIdx.y, l = threadIdx.x;
  int m = l & 15, h = l >> 4;
  const float* base = fgemb + (size_t)b * FP * DD;
  const float* pa = base + (ti * 16 + m) * DD + 2 * h;   // A row fragment ptr
  const float* pb = base + m * DD + 2 * h;               // B tile 0 fragment ptr
  v8f zero = {};
  v8f acc[NFT];
#pragma unroll
  for (int t = 0; t < NFT; ++t) acc[t] = zero;
#pragma unroll 2
  for (int k0 = 0; k0 < DD; k0 += 4) {
    v2f a = *(const v2f*)(pa + k0);                      // loaded once per k
#pragma unroll
    for (int t = 0; t < NFT; ++t) {
      v2f bv = *(const v2f*)(pb + t * 16 * DD + k0);
      acc[t] = wmma4(a, bv, acc[t]);
    }
  }
  float nrow[8], fmrow[8], rowmin[8];
#pragma unroll
  for (int v = 0; v < 8; ++v) {
    int f = ti * 16 + v + 8 * h;
    nrow[v] = fgnrm[b * FP + f];
    fmrow[v] = fgm[b * FP + f];
    rowmin[v] = BIGV;
  }
#pragma unroll
  for (int t = 0; t < NFT; ++t) {
    float ncol  = fgnrm[b * FP + t * 16 + m];
    float cmask = fgm  [b * FP + t * 16 + m];
#pragma unroll
    for (int v = 0; v < 8; ++v) {
      float sim = acc[t][v] / fmaxf(nrow[v] * ncol, EPSV);
      float val = (cmask > 0.f) ? sim : BIGV;
      val = wred_min16(val);
      rowmin[v] = fminf(rowmin[v], val);
    }
  }
  float s = 0.f;
#pragma unroll
  for (int v = 0; v < 8; ++v) s += fmaxf(0.5f - rowmin[v], 0.f) * fmrow[v];
  if ((l & 15) != 0) s = 0.f;
  s = wred_sum32(s);
  if (l == 0) part_ff[b * NFT + ti] = s;
}

// ---- 6: bg-bg gram: row tile x 4 col tiles, k-outer, atomicMin row-mins ----
__global__ void k_bb(const float* __restrict__ emb, const float* __restrict__ nrm,
                     const int* __restrict__ bgidx, const int* __restrict__ nbg,
                     unsigned* __restrict__ bgrmin) {
  int ti = blockIdx.x, cg = blockIdx.y, b = blockIdx.z, l = threadIdx.x;
  int nb = nbg[b];
  if (ti * 16 >= nb || cg * 64 >= nb) return;
  const float* Eb = emb + (size_t)b * DD * TT;
  const int* bg = bgidx + b * TT;
  int m = l & 15, h = l >> 4;
  int rrow = ti * 16 + m;
  const float* pr = Eb + ((rrow < nb) ? bg[rrow] : 0);   // gathered A row
  const float* pcB[4]; float ncol[4]; bool cval[4];
#pragma unroll
  for (int t = 0; t < 4; ++t) {
    int cn = cg * 64 + t * 16 + m;
    cval[t] = cn < nb;
    int ci = cval[t] ? bg[cn] : 0;
    pcB[t] = Eb + ci;
    ncol[t] = cval[t] ? nrm[b * TT + ci] : 1.f;
  }
  v8f zero = {};
  v8f acc[4];
#pragma unroll
  for (int t = 0; t < 4; ++t) acc[t] = zero;
#pragma unroll 2
  for (int k0 = 0; k0 < DD; k0 += 4) {
    int ka = k0 + 2 * h;
    v2f a; a.x = pr[(size_t)ka * TT];                    // loaded once per k,
           a.y = pr[(size_t)(ka + 1) * TT];              // reused by 4 tiles
#pragma unroll
    for (int t = 0; t < 4; ++t) {
      v2f bv; bv.x = pcB[t][(size_t)ka * TT];
              bv.y = pcB[t][(size_t)(ka + 1) * TT];
      acc[t] = wmma4(a, bv, acc[t]);
    }
  }
  float nrow[8], rmin[8];
#pragma unroll
  for (int v = 0; v < 8; ++v) {
    int row = ti * 16 + v + 8 * h;
    nrow[v] = (row < nb) ? nrm[b * TT + bg[row]] : 1.f;
    rmin[v] = BIGV;
  }
#pragma unroll
  for (int t = 0; t < 4; ++t) {
#pragma unroll
    for (int v = 0; v < 8; ++v) {
      float sim = acc[t][v] / fmaxf(nrow[v] * ncol[t], EPSV);
      float val = cval[t] ? sim : BIGV;
      val = wred_min16(val);
      rmin[v] = fminf(rmin[v], val);
    }
  }
  if ((l & 15) == 0) {
#pragma unroll
    for (int v = 0; v < 8; ++v) {
      int row = ti * 16 + v + 8 * h;
      if (row < nb) atomicMin(&bgrmin[b * TT + row], fenc(rmin[v]));
    }
  }
}

// ---- 7: fg x bg: col fragment reused across 9 fg tiles, k-outer ------------
__global__ void k_fb(const float* __restrict__ emb, const float* __restrict__ nrm,
                     const int* __restrict__ bgidx, const int* __restrict__ nbg,
                     const float* __restrict__ fgemb, const float* __restrict__ fgnrm,
                     const float* __restrict__ fgm, unsigned* __restrict__ rowmax,
                     float* __restrict__ part_bf) {
  int tj = blockIdx.x, b = blockIdx.y, l = threadIdx.x;
  int nb = nbg[b];
  if (tj * 16 >= nb) { if (l == 0) part_bf[b * NT + tj] = 0.f; return; }
  const float* Eb = emb + (size_t)b * DD * TT;
  const int* bg = bgidx + b * TT;
  int m = l & 15, h = l >> 4;
  int coln = tj * 16 + m;
  bool cvalid = coln < nb;
  int cidx = cvalid ? bg[coln] : 0;
  float ncol = cvalid ? nrm[b * TT + cidx] : 1.f;
  const float* pc = Eb + cidx;
  __builtin_prefetch(pc, 0, 0);                          // global_prefetch_b8
  const float* paF = fgemb + (size_t)b * FP * DD + m * DD + 2 * h;
  v8f zero = {};
  v8f acc[NFT];
#pragma unroll
  for (int t = 0; t < NFT; ++t) acc[t] = zero;
#pragma unroll 2
  for (int k0 = 0; k0 < DD; k0 += 4) {
    int ka = k0 + 2 * h;
    v2f bcol; bcol.x = pc[(size_t)ka * TT];              // gathered col frag,
              bcol.y = pc[(size_t)(ka + 1) * TT];        // reused by 9 tiles
#pragma unroll
    for (int t = 0; t < NFT; ++t) {
      v2f a = *(const v2f*)(paF + t * 16 * DD + k0);
      acc[t] = wmma4(a, bcol, acc[t]);
    }
  }
  float colmax = -BIGV;
#pragma unroll
  for (int t = 0; t < NFT; ++t) {
#pragma unroll
    for (int v = 0; v < 8; ++v) {
      int f = t * 16 + v + 8 * h;
      float nr = fgnrm[b * FP + f];
      float fm = fgm[b * FP + f];
      float sim = acc[t][v] / fmaxf(nr * ncol, EPSV);
      colmax = fmaxf(colmax, (fm > 0.f) ? sim : -BIGV);
      float rv = cvalid ? sim : -BIGV;
      rv = wred_max16(rv);
      if ((l & 15) == 0) atomicMax(&rowmax[b * FP + f], fenc(rv));
    }
  }
  colmax = fmaxf(colmax, __shfl_xor(colmax, 16, 32));
  float s = (l < 16 && cvalid) ? fmaxf(colmax - 0.1f, 0.f) : 0.f;
  s = wred_sum32(s);
  if (l == 0) part_bf[b * NT + tj] = s;
}

// ---- 8: fixed-order final reduction ----------------------------------------
__global__ void k_final(const int* __restrict__ nbg, const float* __restrict__ fgm,
                        const unsigned* __restrict__ rowmax,
                        const unsigned* __restrict__ bgrmin,
                        const float* __restrict__ part_ff,
                        const float* __restrict__ part_bf, float* out) {
  int l = threadIdx.x;
  float pb = 0.f, hb = 0.f;
  if (l < BB) {
    int b = l;
    int nb = nbg[b];
    float nf = 0.f;
    for (int f = 0; f < FF; ++f) nf += fgm[b * FP + f];
    float sff = 0.f; for (int i = 0; i < NFT; ++i) sff += part_ff[b * NFT + i];
    float sbf = 0.f; for (int i = 0; i < NT;  ++i) sbf += part_bf[b * NT + i];
    float sbb = 0.f;
    for (int r = 0; r < nb; ++r)
      sbb += fmaxf(0.5f - fdec(bgrmin[b * TT + r]), 0.f);
    float sfb = 0.f;
    for (int f = 0; f < FF; ++f)
      sfb += fmaxf(fdec(rowmax[b * FP + f]) - 0.1f, 0.f) * fgm[b * FP + f];
    if (nb > 0) {
      float dn = fmaxf(nf, 1.f), db = fmaxf((float)nb, 1.f);
      pb = sff / dn + sbb / db + sfb / dn + sbf / db;
      hb = 1.f;
    }
  }
  for (int off = 16; off >= 1; off >>= 1) {
    pb += __shfl_xor(pb, off, 32);
    hb += __shfl_xor(hb, off, 32);
  }
  if (l == 0) out[0] = pb / fmaxf(hb, 1.f);
}

extern "C" void kernel_launch(void* const* d_in, const int* in_sizes, int n_in,
                              void* d_out, int out_size, void* d_ws, size_t ws_size,
                              hipStream_t stream) {
  (void)in_sizes; (void)n_in; (void)out_size; (void)ws_size;
  const float* emb    = (const float*)d_in[0];
  const float* cas    = (const float*)d_in[1];
  const int*   click  = (const int*)d_in[2];
  const int*   label  = (const int*)d_in[3];
  const int*   clsnum = (const int*)d_in[4];
  float* out = (float*)d_out;

  char* w = (char*)d_ws;
  size_t o = 0;
  float*    nrm     = (float*)(w + o);    o += (size_t)BB * TT * 4;
  int*      bgidx   = (int*)(w + o);      o += (size_t)BB * TT * 4;
  int*      nbg     = (int*)(w + o);      o += 64;
  int*      fgidx   = (int*)(w + o);      o += (size_t)BB * FP * 4;
  float*    fgnrm   = (float*)(w + o);    o += (size_t)BB * FP * 4;
  float*    fgm     = (float*)(w + o);    o += (size_t)BB * FP * 4;
  unsigned* rowmax  = (unsigned*)(w + o); o += (size_t)BB * FP * 4;
  unsigned* bgrmin  = (unsigned*)(w + o); o += (size_t)BB * TT * 4;
  float*    part_ff = (float*)(w + o);    o += (size_t)BB * NFT * 4;
  float*    part_bf = (float*)(w + o);    o += (size_t)BB * NT * 4;
  float*    fgemb   = (float*)(w + o);    o += (size_t)BB * FP * DD * 4;

  k_init   <<<dim3((BB * TT + 255) / 256), 256, 0, stream>>>(rowmax, bgrmin,
                                                             fgnrm, fgm, fgidx);
  k_norm   <<<dim3(TT / 256, BB), 256, 0, stream>>>(emb, nrm);
  k_compact<<<dim3(BB), 256, 0, stream>>>(click, clsnum, bgidx, nbg);
  k_topk   <<<dim3(CC, BB), 32, 0, stream>>>(cas, label, nrm, fgidx, fgnrm, fgm);
  k_gather <<<dim3(FP, BB), 256, 0, stream>>>(emb, fgidx, fgemb);
  k_ff     <<<dim3(NFT, BB), 32, 0, stream>>>(fgemb, fgnrm, fgm, part_ff);
  k_bb     <<<dim3(NT, NCG, BB), 32, 0, stream>>>(emb, nrm, bgidx, nbg, bgrmin);
  k_fb     <<<dim3(NT, BB), 32, 0, stream>>>(emb, nrm, bgidx, nbg, fgemb, fgnrm,
                                             fgm, rowmax, part_bf);
  k_final  <<<1, 32, 0, stream>>>(nbg, fgm, rowmax, bgrmin, part_ff, part_bf, out);
}